// OR_4544075399223
// MI455X (gfx1250) — compile-verified
//
#include <hip/hip_runtime.h>

// Problem dimensions (fixed by the reference).
#define B_ 128
#define N_ 20000
#define M_ 85400
#define K_ 3

#define TPB 1024       // 32 waves (wave32) per workgroup
#define S_CHUNKS 2     // M-chunks per trajectory -> 2*128 = 256 workgroups

#if defined(__gfx1250__) && __has_builtin(__builtin_amdgcn_global_load_async_to_lds_b128)
#define HAVE_ASYNC_LDS 1
#else
#define HAVE_ASYNC_LDS 0
#endif

#if HAVE_ASYNC_LDS
typedef int v4i __attribute__((ext_vector_type(4)));
typedef __attribute__((address_space(1))) v4i* gv4i_p;  // global (AS1) int4*
typedef __attribute__((address_space(3))) v4i* lv4i_p;  // LDS (AS3) int4*
#endif

__global__ __launch_bounds__(TPB) void or_main(
    const float* __restrict__ v, const float* __restrict__ xl,
    const float* __restrict__ xs, const float* __restrict__ param,
    const float* __restrict__ sign, const int* __restrict__ idx,
    float* __restrict__ outC, float* __restrict__ grad,
    float* __restrict__ outdxl, float* __restrict__ outdxs) {

    // 160000 B of LDS: v[b,:] staging + grad_v[b,:] accumulator (320KB WGP LDS).
    __shared__ __align__(16) float v_sh[N_];
    __shared__ __align__(16) float g_sh[N_];

    const int b = blockIdx.y;
    const int chunk  = (M_ + S_CHUNKS - 1) / S_CHUNKS;
    const int mStart = blockIdx.x * chunk;
    const int mEnd   = (mStart + chunk < M_) ? (mStart + chunk) : M_;
    const int tid    = threadIdx.x;

    const float alpha = param[0], beta = param[1], gamma = param[2];
    const float delta = param[3], eps  = param[4], zeta  = param[5];

    const float* vb = v + (size_t)b * N_;

    // ---- Stage v[b,:] (80KB) into LDS via async global->LDS DMA path ----
#if HAVE_ASYNC_LDS
    for (int i = tid * 4; i < N_; i += TPB * 4) {
        __builtin_amdgcn_global_load_async_to_lds_b128(
            (gv4i_p)(const void*)(vb + i),
            (lv4i_p)(void*)(v_sh + i),
            /*offset=*/0, /*cpol=*/0);
    }
#else
    for (int i = tid; i < N_ / 4; i += TPB)
        ((float4*)v_sh)[i] = ((const float4*)vb)[i];
#endif
    // Zero the LDS gradient accumulator while the async copies are in flight.
    for (int i = tid; i < N_; i += TPB) g_sh[i] = 0.0f;
#if HAVE_ASYNC_LDS
#if __has_builtin(__builtin_amdgcn_s_wait_asynccnt)
    __builtin_amdgcn_s_wait_asynccnt(0);
#else
    asm volatile("s_wait_asynccnt 0x0" ::: "memory");
#endif
#endif
    __syncthreads();

    // ---- Main clause loop: coalesced streams + LDS gathers/atomics ----
    for (int m = mStart + tid; m < mEnd; m += TPB) {
        const size_t bm   = (size_t)b * M_ + m;
        const size_t base = bm * K_;

        int   i0 = __builtin_nontemporal_load(idx + base + 0);
        int   i1 = __builtin_nontemporal_load(idx + base + 1);
        int   i2 = __builtin_nontemporal_load(idx + base + 2);
        float s0 = __builtin_nontemporal_load(sign + base + 0);
        float s1 = __builtin_nontemporal_load(sign + base + 1);
        float s2 = __builtin_nontemporal_load(sign + base + 2);

        float a0 = v_sh[i0] * s0;
        float a1 = v_sh[i1] * s1;
        float a2 = v_sh[i2] * s2;

        // top-2 of 3; strict '>' keeps first-occurrence argmax (top_k tie rule)
        int am; float mx, sec;
        if (a1 > a0) {
            if (a2 > a1) { am = 2; mx = a2; sec = fmaxf(a0, a1); }
            else         { am = 1; mx = a1; sec = fmaxf(a0, a2); }
        } else {
            if (a2 > a0) { am = 2; mx = a2; sec = fmaxf(a0, a1); }
            else         { am = 0; mx = a0; sec = fmaxf(a1, a2); }
        }
        const float C  = (1.0f - mx)  * 0.5f;
        const float Cs = (1.0f - sec) * 0.5f;

        const float xlv = __builtin_nontemporal_load(xl + bm);
        const float xsv = __builtin_nontemporal_load(xs + bm);
        const float gsc = xlv * xsv;                          // G scale
        const float rsc = (1.0f + zeta * xlv) * (1.0f - xsv); // R scale
        const float hotv = Cs * gsc + C * rsc;  // G+R at argmax slot
        const float cold = C * gsc;             // G elsewhere, R=0

        atomicAdd(&g_sh[i0], -((am == 0) ? hotv : cold) * s0);
        atomicAdd(&g_sh[i1], -((am == 1) ? hotv : cold) * s1);
        atomicAdd(&g_sh[i2], -((am == 2) ? hotv : cold) * s2);

        __builtin_nontemporal_store(C, outC + bm);
        __builtin_nontemporal_store(-alpha * (C - delta), outdxl + bm);
        __builtin_nontemporal_store(-beta * (xsv + eps) * (C - gamma), outdxs + bm);
    }
    __syncthreads();

    // ---- Flush this block's 80KB partial into grad_v[b,:] ----
    // (Verified lowering: global_atomic_add_f32 ... scope:SCOPE_DEV)
    float* gb = grad + (size_t)b * N_;
    for (int i = tid; i < N_; i += TPB) {
        __hip_atomic_fetch_add(&gb[i], g_sh[i],
                               __ATOMIC_RELAXED, __HIP_MEMORY_SCOPE_AGENT);
    }
}

__global__ __launch_bounds__(TPB) void or_zero_grad(float* __restrict__ grad) {
    size_t i = (size_t)blockIdx.x * blockDim.x + threadIdx.x;
    if (i < (size_t)B_ * N_) grad[i] = 0.0f;
}

extern "C" void kernel_launch(void* const* d_in, const int* in_sizes, int n_in,
                              void* d_out, int out_size, void* d_ws, size_t ws_size,
                              hipStream_t stream) {
    const float* v     = (const float*)d_in[0];
    const float* xl    = (const float*)d_in[1];
    const float* xs    = (const float*)d_in[2];
    const float* param = (const float*)d_in[3];
    const float* sgn   = (const float*)d_in[4];
    const int*   idx   = (const int*)d_in[5];

    float* out    = (float*)d_out;
    float* outC   = out;                                  // (B,M)
    float* grad   = outC + (size_t)B_ * M_;               // (B,N)
    float* outdxl = grad + (size_t)B_ * N_;               // (B,M)
    float* outdxs = outdxl + (size_t)B_ * M_;             // (B,M)

    const size_t ztotal = (size_t)B_ * N_;
    or_zero_grad<<<dim3((unsigned)((ztotal + TPB - 1) / TPB)), dim3(TPB), 0, stream>>>(grad);

    or_main<<<dim3(S_CHUNKS, B_), dim3(TPB), 0, stream>>>(
        v, xl, xs, param, sgn, idx, outC, grad, outdxl, outdxs);
}